// SignedConv_10660108829350
// MI455X (gfx1250) — compile-verified
//
#include <hip/hip_runtime.h>

// ---------------------------------------------------------------------------
// SignedConv (first_aggr=True) for MI455X / gfx1250, wave32.
//   N=50000 nodes, F=256 in-feats, OUT=128, E=800000 edges per sign.
// Phases: fp32-WMMA GEMMs (att + output), L2-resident edge gather dots,
//         atomic scatter-mean. All matrix math via V_WMMA_F32_16X16X4_F32.
// ---------------------------------------------------------------------------

typedef __attribute__((ext_vector_type(2))) float v2f;
typedef __attribute__((ext_vector_type(8))) float v8f;

#define NNODES 50000
#define FEAT   256
#define OUTF   128
#define NEDGE  800000

// ---------------------------------------------------------------------------
// Y[M x P] = X[M x 256] @ W^T  (+ bias per column).  W stored [P x 256] rowmaj.
// One wave -> one 16x16 tile, K loop in steps of 4 via v_wmma_f32_16x16x4_f32.
// A frag (16x4 fp32): lanes 0-15 hold K={k,k+1} of row M=lane,
//                     lanes 16-31 hold K={k+2,k+3} of row M=lane-16.
// B frag (4x16) mirrored; D: VGPR i = row m0+i (lanes<16) / m0+8+i (lanes>=16).
// ---------------------------------------------------------------------------
__global__ __launch_bounds__(256) void gemm_bias_wmma(
    const float* __restrict__ X, const float* __restrict__ W,
    const float* __restrict__ B, float* __restrict__ Y, int P)
{
  const int lane  = threadIdx.x & 31;
  const int wave  = threadIdx.x >> 5;
  const int m0    = blockIdx.x << 4;
  const int n0    = ((blockIdx.y << 3) + wave) << 4;
  const int lmod  = lane & 15;
  const int lhalf = lane >> 4;
  const int koff  = lhalf << 1;

  const float* xrow = X + (size_t)(m0 + lmod) * FEAT + koff;
  const float* wrow = W + (size_t)(n0 + lmod) * FEAT + koff;

  v8f acc = {0.f, 0.f, 0.f, 0.f, 0.f, 0.f, 0.f, 0.f};
#pragma unroll 8
  for (int k = 0; k < FEAT; k += 4) {
    v2f a = *(const v2f*)(xrow + k);
    v2f b = *(const v2f*)(wrow + k);
    acc = __builtin_amdgcn_wmma_f32_16x16x4_f32(
        false, a, false, b, (short)0, acc, false, false);
  }

  const float bias = B ? B[n0 + lmod] : 0.0f;
  float* yp = Y + (size_t)(m0 + (lhalf << 3)) * P + (n0 + lmod);
#pragma unroll
  for (int i = 0; i < 8; ++i) yp[(size_t)i * P] = acc[i] + bias;
}

// ---------------------------------------------------------------------------
// out[m, colOff+o] = (S[m,:]*inv[m]) @ W1^T + X[m,:] @ W2^T + B2[o]
// Same WMMA tiling; two K loops share one fp32 accumulator (exact fusion of
// the aggregated-term GEMM and the self-loop GEMM). OUT row stride = 2*OUTF.
// ---------------------------------------------------------------------------
__global__ __launch_bounds__(256) void gemm_out_wmma(
    const float* __restrict__ S, const float* __restrict__ INV,
    const float* __restrict__ X,
    const float* __restrict__ W1, const float* __restrict__ W2,
    const float* __restrict__ B2, float* __restrict__ OUT, int colOff)
{
  const int lane  = threadIdx.x & 31;
  const int wave  = threadIdx.x >> 5;
  const int m0    = blockIdx.x << 4;
  const int n0    = wave << 4;            // 8 waves cover 128 output cols
  const int lmod  = lane & 15;
  const int lhalf = lane >> 4;
  const int koff  = lhalf << 1;
  const int arow  = m0 + lmod;

  const float scale = INV[arow];          // 1 / max(count, 1)
  const float* srow  = S  + (size_t)arow * FEAT + koff;
  const float* xrow  = X  + (size_t)arow * FEAT + koff;
  const float* w1row = W1 + (size_t)(n0 + lmod) * FEAT + koff;
  const float* w2row = W2 + (size_t)(n0 + lmod) * FEAT + koff;

  v8f acc = {0.f, 0.f, 0.f, 0.f, 0.f, 0.f, 0.f, 0.f};
#pragma unroll 8
  for (int k = 0; k < FEAT; k += 4) {     // aggregated (weighted-mean) term
    v2f a = *(const v2f*)(srow + k) * scale;
    v2f b = *(const v2f*)(w1row + k);
    acc = __builtin_amdgcn_wmma_f32_16x16x4_f32(
        false, a, false, b, (short)0, acc, false, false);
  }
#pragma unroll 8
  for (int k = 0; k < FEAT; k += 4) {     // self-loop (cc) term
    v2f a = *(const v2f*)(xrow + k);
    v2f b = *(const v2f*)(w2row + k);
    acc = __builtin_amdgcn_wmma_f32_16x16x4_f32(
        false, a, false, b, (short)0, acc, false, false);
  }

  const float bias = B2[n0 + lmod];
  float* yp = OUT + (size_t)(m0 + (lhalf << 3)) * (2 * OUTF) + colOff + (n0 + lmod);
#pragma unroll
  for (int i = 0; i < 8; ++i) yp[(size_t)i * (2 * OUTF)] = acc[i] + bias;
}

// ---------------------------------------------------------------------------
// w[e] = dot(att[r[e]], x[l[e]]) over 256 dims. One wave per edge; each lane
// covers 8 contiguous floats (2x float4, b128 loads hitting L2), then a
// wave32 xor-shuffle reduction.
// ---------------------------------------------------------------------------
__global__ __launch_bounds__(256) void edge_score(
    const int* __restrict__ EI, const float* __restrict__ ATT,
    const float* __restrict__ X, float* __restrict__ Wout, int E)
{
  const int lane = threadIdx.x & 31;
  const int wave = threadIdx.x >> 5;
  const int e = (blockIdx.x << 3) + wave;
  if (e >= E) return;
  const int r = EI[e];
  const int l = EI[E + e];

  const float4* ap = (const float4*)(ATT + (size_t)r * FEAT) + (lane << 1);
  const float4* xp = (const float4*)(X   + (size_t)l * FEAT) + (lane << 1);
  float4 a0 = ap[0], a1 = ap[1];
  float4 x0 = xp[0], x1 = xp[1];
  float v = a0.x * x0.x + a0.y * x0.y + a0.z * x0.z + a0.w * x0.w
          + a1.x * x1.x + a1.y * x1.y + a1.z * x1.z + a1.w * x1.w;
#pragma unroll
  for (int off = 16; off >= 1; off >>= 1) v += __shfl_xor(v, off, 32);
  if (lane == 0) Wout[e] = v;
}

// ---------------------------------------------------------------------------
// Scatter: S[r[e], :] += w[e] * x[l[e], :],  CNT[r[e]] += 1.
// One wave per edge; 8 f32 atomics per lane (global_atomic_add_f32, L2-resident
// accumulator). Count once per edge (lane 0).
// ---------------------------------------------------------------------------
__global__ __launch_bounds__(256) void edge_scatter(
    const int* __restrict__ EI, const float* __restrict__ Wv,
    const float* __restrict__ X, float* __restrict__ S,
    float* __restrict__ CNT, int E)
{
  const int lane = threadIdx.x & 31;
  const int wave = threadIdx.x >> 5;
  const int e = (blockIdx.x << 3) + wave;
  if (e >= E) return;
  const int r = EI[e];
  const int l = EI[E + e];
  const float w = Wv[e];

  const float4* xp = (const float4*)(X + (size_t)l * FEAT) + (lane << 1);
  float4 x0 = xp[0], x1 = xp[1];
  float* sp = S + (size_t)r * FEAT + (lane << 3);
  atomicAdd(sp + 0, w * x0.x);
  atomicAdd(sp + 1, w * x0.y);
  atomicAdd(sp + 2, w * x0.z);
  atomicAdd(sp + 3, w * x0.w);
  atomicAdd(sp + 4, w * x1.x);
  atomicAdd(sp + 5, w * x1.y);
  atomicAdd(sp + 6, w * x1.z);
  atomicAdd(sp + 7, w * x1.w);
  if (lane == 0) atomicAdd(CNT + r, 1.0f);
}

__global__ __launch_bounds__(256) void finalize_inv(float* __restrict__ CNT, int n)
{
  int i = blockIdx.x * blockDim.x + threadIdx.x;
  if (i < n) {
    float c = CNT[i];
    CNT[i] = 1.0f / fmaxf(c, 1.0f);
  }
}

// ---------------------------------------------------------------------------
extern "C" void kernel_launch(void* const* d_in, const int* in_sizes, int n_in,
                              void* d_out, int out_size, void* d_ws, size_t ws_size,
                              hipStream_t stream)
{
  const float* x        = (const float*)d_in[0];
  const int*   pos_ei   = (const int*)d_in[1];
  const int*   neg_ei   = (const int*)d_in[2];
  const float* W_pos    = (const float*)d_in[3];
  const float* W_pos_cc = (const float*)d_in[4];
  const float* b_pos_cc = (const float*)d_in[5];
  const float* W_neg    = (const float*)d_in[6];
  const float* W_neg_cc = (const float*)d_in[7];
  const float* b_neg_cc = (const float*)d_in[8];
  const float* W_patt   = (const float*)d_in[9];
  const float* b_patt   = (const float*)d_in[10];
  const float* W_natt   = (const float*)d_in[11];
  const float* b_natt   = (const float*)d_in[12];
  float* out = (float*)d_out;

  // Workspace: [S/att: N*256] [edge scores: E] [counts: N]  (~55 MB fp32)
  float* Sbuf = (float*)d_ws;                       // att matrix, then scatter sum
  float* Wbuf = Sbuf + (size_t)NNODES * FEAT;       // per-edge scores
  float* Cbuf = Wbuf + (size_t)NEDGE;               // counts -> inverse counts

  const dim3 blk(256);
  const dim3 gAtt(NNODES / 16, FEAT / 128);         // 3125 x 2 (8 waves = 128 cols)
  const dim3 gOut(NNODES / 16, 1);                  // 3125   (8 waves = 128 cols)
  const int  gEdge = (NEDGE + 7) / 8;               // 8 edges (waves) per block
  const int  gFin  = (NNODES + 255) / 256;

  const int*   ei[2]   = { pos_ei, neg_ei };
  const float* Watt[2] = { W_patt, W_natt };
  const float* batt[2] = { b_patt, b_natt };
  const float* Wagg[2] = { W_pos,  W_neg  };
  const float* Wcc[2]  = { W_pos_cc, W_neg_cc };
  const float* bcc[2]  = { b_pos_cc, b_neg_cc };

  for (int s = 0; s < 2; ++s) {
    // 1) attention matrix: att = x @ W_att^T + b_att    [N, 256]
    gemm_bias_wmma<<<gAtt, blk, 0, stream>>>(x, Watt[s], batt[s], Sbuf, FEAT);
    // 2) per-edge scores: w[e] = dot(att[r], x[l])
    edge_score<<<gEdge, blk, 0, stream>>>(ei[s], Sbuf, x, Wbuf, NEDGE);
    // 3) reset accumulators (att no longer needed; buffer reused as sum)
    hipMemsetAsync(Sbuf, 0, (size_t)NNODES * FEAT * sizeof(float), stream);
    hipMemsetAsync(Cbuf, 0, (size_t)NNODES * sizeof(float), stream);
    // 4) weighted scatter-sum + counts
    edge_scatter<<<gEdge, blk, 0, stream>>>(ei[s], Wbuf, x, Sbuf, Cbuf, NEDGE);
    // 5) counts -> 1/max(cnt,1)
    finalize_inv<<<gFin, blk, 0, stream>>>(Cbuf, NNODES);
    // 6) out[:, s*128 : s*128+128] = mean @ W^T + x @ Wcc^T + bcc
    gemm_out_wmma<<<gOut, blk, 0, stream>>>(Sbuf, Cbuf, x, Wagg[s], Wcc[s],
                                            bcc[s], out, s * OUTF);
  }
}